// SS2D_10170482557730
// MI455X (gfx1250) — compile-verified
//
#include <hip/hip_runtime.h>
#include <hip/hip_bf16.h>
#include <math.h>

// ---------------- problem constants ----------------
#define Bn    4
#define Hh    64
#define Wd    64
#define Ldim  4096      // H*W
#define DM    96        // d_model
#define Ee    192       // d_inner
#define Nst   16        // d_state
#define Rr    6         // dt_rank
#define Kk    4         // scan directions
#define CPROJ 38        // R + 2N

typedef __attribute__((ext_vector_type(2))) float v2f;
typedef __attribute__((ext_vector_type(8))) float v8f;

__device__ __forceinline__ float siluf(float x) { return x / (1.0f + __expf(-x)); }

// source index into the hw-ordered (H*W) plane for scan direction k at scan pos l
__device__ __forceinline__ int scan_map(int k, int l) {
  int ll = (k >= 2) ? (Ldim - 1 - l) : l;
  if (k & 1) ll = ((ll & 63) << 6) | (ll >> 6);   // transpose (w,h)->(h,w)
  return ll;
}

// ================= 1) in_proj GEMM: xz = X(16384x96) * W^T(96x384) =================
// fused: cols [0,192) -> xx in (B,E,L) channel-first;  cols [192,384) -> silu -> z (B,L,E)
__global__ void k_inproj(const float* __restrict__ X, const float* __restrict__ W,
                         float* __restrict__ xx, float* __restrict__ z) {
  const int wave = (blockIdx.x * blockDim.x + threadIdx.x) >> 5;
  const int lane = threadIdx.x & 31;
  const int half = lane >> 4, lid = lane & 15;
  const int NT = (2 * Ee) / 16;                 // 24 col tiles
  const int m0 = (wave / NT) * 16;
  const int n0 = (wave % NT) * 16;

  const float* arow = X + (size_t)(m0 + lid) * DM;   // A row (per lane)
  const float* bcol = W + (size_t)(n0 + lid) * DM;   // B^T col == W row (per lane)

  v8f acc = {};
  for (int k0 = 0; k0 < DM; k0 += 4) {
    const int kk = k0 + 2 * half;
    v2f a; a.x = arow[kk]; a.y = arow[kk + 1];
    v2f b; b.x = bcol[kk]; b.y = bcol[kk + 1];
    acc = __builtin_amdgcn_wmma_f32_16x16x4_f32(false, a, false, b, (short)0, acc,
                                                false, false);
  }
  const int n = n0 + lid;
#pragma unroll
  for (int v = 0; v < 8; ++v) {
    const int m = m0 + v + 8 * half;
    const int bb = m >> 12;            // m / 4096
    const int l  = m & 4095;
    const float val = acc[v];
    if (n < Ee) xx[((size_t)bb * Ee + n) * Ldim + l] = val;            // (B,E,L)
    else        z[(size_t)m * Ee + (n - Ee)] = siluf(val);             // (B,L,E)
  }
}

// ================= 2) depthwise 3x3 conv + SiLU (zero pad) =================
__global__ void k_conv(const float* __restrict__ xx, const float* __restrict__ cw,
                       const float* __restrict__ cb, float* __restrict__ xc) {
  const int idx = blockIdx.x * blockDim.x + threadIdx.x;   // over B*E*L
  const int l = idx & 4095;
  const int be = idx >> 12;
  const int e = be % Ee;
  const int h = l >> 6, w = l & 63;
  const float* p = xx + (size_t)be * Ldim;
  const float* wt = cw + e * 9;
  float s = cb[e];
#pragma unroll
  for (int dy = -1; dy <= 1; ++dy)
#pragma unroll
    for (int dx = -1; dx <= 1; ++dx) {
      const int hh = h + dy, ww = w + dx;
      if (hh >= 0 && hh < 64 && ww >= 0 && ww < 64)
        s += p[hh * 64 + ww] * wt[(dy + 1) * 3 + (dx + 1)];
    }
  xc[idx] = siluf(s);
}

// ================= 3) x_proj GEMM per (b,k): xdbl[c,l] = xpw[k] (38x192) * xs (192x4096)
// cross-scan gather folded into B-operand load; M padded 38 -> 48 (3 tiles)
__global__ void k_xproj(const float* __restrict__ xc, const float* __restrict__ xpw,
                        float* __restrict__ xdbl) {
  const int wave = (blockIdx.x * blockDim.x + threadIdx.x) >> 5;
  const int lane = threadIdx.x & 31;
  const int half = lane >> 4, lid = lane & 15;
  const int nt = wave & 255;                    // 256 l-tiles
  const int rest = wave >> 8;
  const int mt = rest % 3;                      // 3 c-tiles (rows 0..47)
  const int bk = rest / 3;                      // 16 (b,k) pairs
  const int b = bk >> 2, k = bk & 3;

  const int c_lane = mt * 16 + lid;
  const bool avalid = (c_lane < CPROJ);
  const float* arow = xpw + ((size_t)k * CPROJ + (avalid ? c_lane : 0)) * Ee;
  const int l_lane = nt * 16 + lid;
  const float* bbase = xc + (size_t)b * Ee * Ldim + scan_map(k, l_lane);

  v8f acc = {};
  for (int e0 = 0; e0 < Ee; e0 += 4) {
    const int e = e0 + 2 * half;
    v2f a;
    a.x = avalid ? arow[e] : 0.0f;
    a.y = avalid ? arow[e + 1] : 0.0f;
    v2f bf;
    bf.x = bbase[(size_t)e * Ldim];
    bf.y = bbase[(size_t)(e + 1) * Ldim];
    acc = __builtin_amdgcn_wmma_f32_16x16x4_f32(false, a, false, bf, (short)0, acc,
                                                false, false);
  }
  const int l = nt * 16 + lid;
#pragma unroll
  for (int v = 0; v < 8; ++v) {
    const int c = mt * 16 + v + 8 * half;
    if (c < CPROJ) xdbl[((size_t)bk * CPROJ + c) * Ldim + l] = acc[v];
  }
}

// ================= 4) selective scan: block per (b,k), thread per e =================
__global__ void k_scan(const float* __restrict__ xc, const float* __restrict__ xdbl,
                       const float* __restrict__ dtw, const float* __restrict__ dtb,
                       const float* __restrict__ alog, const float* __restrict__ Dsp,
                       float* __restrict__ oy) {
  const int bk = blockIdx.x;            // 0..15
  const int b = bk >> 2, k = bk & 3;
  const int e = threadIdx.x;            // 0..191
  const int ke = k * Ee + e;

  float dtp[Rr];
#pragma unroll
  for (int r = 0; r < Rr; ++r) dtp[r] = dtw[(size_t)ke * Rr + r];
  const float bias = dtb[ke];
  float Aa[Nst];
#pragma unroll
  for (int n = 0; n < Nst; ++n) Aa[n] = -__expf(alog[(size_t)ke * Nst + n]);
  const float Dv = Dsp[ke];
  float h[Nst];
#pragma unroll
  for (int n = 0; n < Nst; ++n) h[n] = 0.0f;

  __shared__ float sh[CPROJ][64];       // dt rows 0..5, B rows 6..21, C rows 22..37
  const float* src = xdbl + (size_t)bk * CPROJ * Ldim;
  const float* up = xc + (size_t)(b * Ee + e) * Ldim;
  float* outp = oy + (size_t)(bk * Ee + e) * Ldim;

  for (int l0 = 0; l0 < Ldim; l0 += 64) {
    __syncthreads();
    for (int i = threadIdx.x; i < CPROJ * 64; i += blockDim.x) {
      const int c = i >> 6, j = i & 63;
      sh[c][j] = src[(size_t)c * Ldim + l0 + j];
    }
    __syncthreads();
    for (int j = 0; j < 64; ++j) {
      const int l = l0 + j;
      const float u = up[scan_map(k, l)];
      float dt = bias;
#pragma unroll
      for (int r = 0; r < Rr; ++r) dt += dtp[r] * sh[r][j];
      const float d = (dt > 20.0f) ? dt : log1pf(__expf(dt));   // softplus
      const float du = d * u;
      float y = 0.0f;
#pragma unroll
      for (int n = 0; n < Nst; ++n) {
        h[n] = __expf(d * Aa[n]) * h[n] + du * sh[Rr + n][j];
        y += h[n] * sh[Rr + Nst + n][j];
      }
      outp[l] = y + Dv * u;
    }
  }
}

// ================= 5) cross-merge + LayerNorm(E) + z-gate =================
__global__ void k_merge(const float* __restrict__ oy, const float* __restrict__ z,
                        const float* __restrict__ gam, const float* __restrict__ bet,
                        float* __restrict__ yg) {
  const int m = blockIdx.x;             // 0..B*L-1
  const int e = threadIdx.x;            // 0..191
  const int b = m >> 12, l = m & 4095;
  const int hh = l >> 6, ww = l & 63;
  const int lT = ww * 64 + hh;
  const size_t base = (size_t)b * Kk * Ee * Ldim;
  const size_t pe = (size_t)e * Ldim;
  const size_t plane = (size_t)Ee * Ldim;

  float v = oy[base + 0 * plane + pe + l]
          + oy[base + 2 * plane + pe + (Ldim - 1 - l)]
          + oy[base + 1 * plane + pe + lT]
          + oy[base + 3 * plane + pe + (Ldim - 1 - lT)];

  __shared__ float s1[Ee], s2[Ee];
  s1[e] = v; s2[e] = v * v;
  __syncthreads();
  if (e < 64) { s1[e] += s1[e + 128]; s2[e] += s2[e + 128]; }
  __syncthreads();
  if (e < 64) { s1[e] += s1[e + 64];  s2[e] += s2[e + 64];  }
  __syncthreads();
  for (int s = 32; s > 0; s >>= 1) {
    if (e < s) { s1[e] += s1[e + s]; s2[e] += s2[e + s]; }
    __syncthreads();
  }
  const float mu = s1[0] * (1.0f / Ee);
  const float var = s2[0] * (1.0f / Ee) - mu * mu;
  const float yn = (v - mu) * rsqrtf(var + 1e-5f) * gam[e] + bet[e];
  yg[(size_t)m * Ee + e] = yn * z[(size_t)m * Ee + e];
}

// ================= 6) out_proj GEMM: out(16384x96) = yg(16384x192) * W^T =================
__global__ void k_outproj(const float* __restrict__ yg, const float* __restrict__ W,
                          float* __restrict__ out) {
  const int wave = (blockIdx.x * blockDim.x + threadIdx.x) >> 5;
  const int lane = threadIdx.x & 31;
  const int half = lane >> 4, lid = lane & 15;
  const int NT = DM / 16;                        // 6 col tiles
  const int m0 = (wave / NT) * 16;
  const int n0 = (wave % NT) * 16;

  const float* arow = yg + (size_t)(m0 + lid) * Ee;
  const float* bcol = W + (size_t)(n0 + lid) * Ee;

  v8f acc = {};
  for (int k0 = 0; k0 < Ee; k0 += 4) {
    const int kk = k0 + 2 * half;
    v2f a; a.x = arow[kk]; a.y = arow[kk + 1];
    v2f bf; bf.x = bcol[kk]; bf.y = bcol[kk + 1];
    acc = __builtin_amdgcn_wmma_f32_16x16x4_f32(false, a, false, bf, (short)0, acc,
                                                false, false);
  }
  const int n = n0 + lid;
#pragma unroll
  for (int v = 0; v < 8; ++v) {
    const int m = m0 + v + 8 * half;
    out[(size_t)m * DM + n] = acc[v];
  }
}

// ================= launch =================
extern "C" void kernel_launch(void* const* d_in, const int* in_sizes, int n_in,
                              void* d_out, int out_size, void* d_ws, size_t ws_size,
                              hipStream_t stream) {
  const float* x    = (const float*)d_in[0];   // (B,H,W,96)
  const float* ipw  = (const float*)d_in[1];   // (384,96)
  const float* cw   = (const float*)d_in[2];   // (192,1,3,3)
  const float* cb   = (const float*)d_in[3];   // (192,)
  const float* xpw  = (const float*)d_in[4];   // (4,38,192)
  const float* dtw  = (const float*)d_in[5];   // (4,192,6)
  const float* dtb  = (const float*)d_in[6];   // (4,192)
  const float* alog = (const float*)d_in[7];   // (768,16)
  const float* Dsp  = (const float*)d_in[8];   // (768,)
  const float* lg   = (const float*)d_in[9];   // (192,)
  const float* lb   = (const float*)d_in[10];  // (192,)
  const float* opw  = (const float*)d_in[11];  // (96,192)

  float* ws = (float*)d_ws;
  size_t o = 0;
  float* xx  = ws + o; o += (size_t)Bn * Ee * Ldim;          // (B,E,L)
  float* z   = ws + o; o += (size_t)Bn * Ldim * Ee;          // (B,L,E)
  float* xc  = ws + o; o += (size_t)Bn * Ee * Ldim;          // (B,E,L)
  float* xd  = ws + o; o += (size_t)Bn * Kk * CPROJ * Ldim;  // (B,K,38,L)
  float* oy  = ws + o; o += (size_t)Bn * Kk * Ee * Ldim;     // (B,K,E,L)
  float* yg  = ws + o; o += (size_t)Bn * Ldim * Ee;          // (B,L,E)
  float* out = (float*)d_out;

  // 1024 m-tiles * 24 n-tiles = 24576 waves = 3072 blocks of 8 waves
  k_inproj<<<3072, 256, 0, stream>>>(x, ipw, xx, z);
  k_conv<<<(Bn * Ee * Ldim) / 256, 256, 0, stream>>>(xx, cw, cb, xc);
  // 16 (b,k) * 3 c-tiles * 256 l-tiles = 12288 waves = 1536 blocks
  k_xproj<<<1536, 256, 0, stream>>>(xc, xpw, xd);
  k_scan<<<Bn * Kk, Ee, 0, stream>>>(xc, xd, dtw, dtb, alog, Dsp, oy);
  k_merge<<<Bn * Ldim, Ee, 0, stream>>>(oy, z, lg, lb, yg);
  // 1024 m-tiles * 6 n-tiles = 6144 waves = 768 blocks
  k_outproj<<<768, 256, 0, stream>>>(yg, opw, out);
}